// EventImageEncoder_39444979646671
// MI455X (gfx1250) — compile-verified
//
#include <hip/hip_runtime.h>
#include <stdint.h>

// ---------------------------------------------------------------------------
// Event-image encoder for MI455X (gfx1250, wave32).
//
//   kernel 1: out = event_image                (2.4 MB copy, trivial)
//   kernel 2: stream 16.7M events (268 MB) chunk-wise into LDS using the
//             CDNA5 Tensor Data Mover (tensor_load_to_lds, TENSORcnt),
//             double-buffered (2 x 32 KB), then each lane reads one event
//             (float4, ds_load_b128) and issues one global_atomic_add_f32
//             into the L2-resident 2x480x640 output.
//
// Roofline: 268 MB streamed @ 23.3 TB/s ≈ 11.5 µs floor; output (2.4 MB)
// lives in the 192 MB L2, so the 16.7M fp32 atomics are L2-resident and
// fire-and-forget (STOREcnt). TDM overlaps the global streaming with the
// atomic-issue phase without burning VGPR load slots; wave 0 builds the D#
// (SALU only) while the other 7 waves wait at the barrier; all 8 consume each chunk.
// ---------------------------------------------------------------------------

#define SENSOR_W 640
#define SENSOR_H 480
#define HW_PIX   (SENSOR_W * SENSOR_H)

#define BLOCK_THREADS 256          // 8 waves on wave32
#define CHUNK_EVENTS  2048         // 32 KB per buffer (float4 per event)
#define CHUNK_DW      (CHUNK_EVENTS * 4)
#define EV_PER_THREAD (CHUNK_EVENTS / BLOCK_THREADS)

typedef uint32_t u32;
typedef __attribute__((ext_vector_type(4))) u32 u32x4;
typedef __attribute__((ext_vector_type(8))) int i32x8;
typedef __attribute__((ext_vector_type(4))) int i32x4;

#if defined(__HIP_DEVICE_COMPILE__) && __has_builtin(__builtin_amdgcn_tensor_load_to_lds) && __has_builtin(__builtin_amdgcn_s_wait_tensorcnt)
#define USE_TDM 1
#else
#define USE_TDM 0
#endif

__global__ void ei_init_out(const float* __restrict__ img,
                            float* __restrict__ out, int n) {
  int i = blockIdx.x * blockDim.x + threadIdx.x;
  if (i < n) out[i] = img[i];
}

__global__ __launch_bounds__(BLOCK_THREADS)
void ei_hist(const float* __restrict__ events,   // [n_events,4] = x,y,t,p
             float* __restrict__ out,            // [2,H,W]
             int n_events, int n_chunks) {
  __shared__ float4 buf[2][CHUNK_EVENTS];        // 2 x 32 KB double buffer

  const int tid = threadIdx.x;

#if USE_TDM
  // Wave 0 owns DMA issue; condition is uniform per wave (wave32).
  const bool issuer = (tid < 32);

  // Build a Tensor DMA Descriptor (D#) for one 1-D chunk and fire the TDM.
  //   group0: count=1 | lds_addr | global_addr[56:0] | type=2
  //   group1: data_size=4B, tensor_dim0 = remaining dwords (HW zero-fills OOB),
  //           tile_dim0 = CHUNK_DW, 1-D tile (tile_dim1=0)
  auto tdm_issue = [&](int b, long long base_ev) {
    unsigned long long gaddr =
        (unsigned long long)(uintptr_t)(events + base_ev * 4);
    // Flat LDS pointer: low 32 bits are the LDS byte address (aperture in MSBs).
    u32 lds_addr = (u32)(uintptr_t)(&buf[b][0]);
    long long rem_dw = ((long long)n_events - base_ev) * 4;
    u32 tdim0 = (rem_dw > (long long)CHUNK_DW) ? (u32)CHUNK_DW : (u32)rem_dw;

    u32x4 g0;
    g0.x = 1u;                                            // count=1 (valid)
    g0.y = lds_addr;                                      // lds_addr[31:0]
    g0.z = (u32)(gaddr & 0xFFFFFFFFu);                    // global_addr[31:0]
    g0.w = (u32)((gaddr >> 32) & 0x01FFFFFFu) | (2u << 30); // [56:32] | type=2

    i32x8 g1;
    g1[0] = (int)(2u << 16);                    // workgroup_mask=0, data_size=4B
    g1[1] = (int)((tdim0 & 0xFFFFu) << 16);     // tensor_dim0[15:0]
    g1[2] = (int)((tdim0 >> 16) | (1u << 16));  // tensor_dim0[31:16], tensor_dim1=1
    g1[3] = (int)((u32)CHUNK_DW << 16);         // tile_dim0 = 8192 dwords
    g1[4] = 0;                                  // tile_dim1=0, tile_dim2=0 (1-D)
    g1[5] = (int)tdim0;                         // tensor_dim0_stride[31:0]
    g1[6] = 0;                                  // stride hi / dim1_stride lo
    g1[7] = 0;

    i32x4 z4 = {0, 0, 0, 0};                    // groups 2/3 unused (<=2D)
    i32x8 z8 = {0, 0, 0, 0, 0, 0, 0, 0};        // extra arg (6-arg toolchain)
    __builtin_amdgcn_tensor_load_to_lds(g0, g1, z4, z4, z8, 0);
  };
#endif

  const long long chunk_stride = (long long)gridDim.x;
  long long c = (long long)blockIdx.x;
  int cur = 0;

#if USE_TDM
  if (c < n_chunks && issuer) tdm_issue(0, c * CHUNK_EVENTS);
#endif

  for (; c < n_chunks; c += chunk_stride) {
    const long long base = c * CHUNK_EVENTS;
    const long long next_c = c + chunk_stride;

#if USE_TDM
    if (issuer) {
      if (next_c < n_chunks) {
        // Prefetch next chunk into the other buffer, then wait for current.
        tdm_issue(cur ^ 1, next_c * CHUNK_EVENTS);
        __builtin_amdgcn_s_wait_tensorcnt(1);
      } else {
        __builtin_amdgcn_s_wait_tensorcnt(0);
      }
    }
#else
    // Fallback: cooperative global->LDS staging (coalesced b128 loads).
    #pragma unroll
    for (int k = 0; k < EV_PER_THREAD; ++k) {
      long long e = base + tid + k * BLOCK_THREADS;
      if (e < n_events)
        buf[cur][tid + k * BLOCK_THREADS] =
            reinterpret_cast<const float4*>(events)[e];
    }
#endif
    __syncthreads();   // LDS chunk `cur` visible to all 8 waves

    #pragma unroll
    for (int k = 0; k < EV_PER_THREAD; ++k) {
      const int idx = tid + k * BLOCK_THREADS;
      if (base + idx < n_events) {
        float4 ev = buf[cur][idx];               // ds_load_b128
        int x = (int)ev.x;                       // trunc == astype(int32) here
        int y = (int)ev.y;
        if (x >= 0 && x < SENSOR_W && y >= 0 && y < SENSOR_H) {
          int flat = y * SENSOR_W + x;
          int ch   = (ev.w > 0.0f) ? 0 : 1;      // polarity -> channel
          // HW fp32 atomic (no return): global_atomic_add_f32, L2-resident.
          unsafeAtomicAdd(&out[ch * HW_PIX + flat], 1.0f);
        }
      }
    }
    __syncthreads();   // all waves done reading before buffer is re-filled
    cur ^= 1;
  }
}

extern "C" void kernel_launch(void* const* d_in, const int* in_sizes, int n_in,
                              void* d_out, int out_size, void* d_ws, size_t ws_size,
                              hipStream_t stream) {
  const float* events = (const float*)d_in[0];   // [N,4] float32
  const float* img    = (const float*)d_in[1];   // [2,H,W] float32
  float* out          = (float*)d_out;           // [2,H,W] float32

  const int n_events = in_sizes[0] / 4;

  // 1) out = event_image
  if (out_size > 0) {
    int g = (out_size + 255) / 256;
    ei_init_out<<<g, 256, 0, stream>>>(img, out, out_size);
  }

  // 2) scatter-add events
  const int n_chunks = (n_events + CHUNK_EVENTS - 1) / CHUNK_EVENTS;
  if (n_chunks > 0) {
    int grid = n_chunks < 2048 ? n_chunks : 2048;  // ~16K waves available
    ei_hist<<<grid, BLOCK_THREADS, 0, stream>>>(events, out, n_events, n_chunks);
  }
}